// SpatialTransformer_10299331576022
// MI455X (gfx1250) — compile-verified
//
#include <hip/hip_runtime.h>
#include <stdint.h>

// ---------------- problem constants ----------------
#define OUT_HW          400
#define IMG_HW          512
#define CH              3
#define ROW_DW          (IMG_HW * CH)            // 1536 dwords per image row
#define IMG_DW          (IMG_HW * IMG_HW * CH)   // dwords per batch image
#define TILE            16
#define TILES_PER_SIDE  (OUT_HW / TILE)          // 25
#define TILES_PER_BATCH (TILES_PER_SIDE * TILES_PER_SIDE) // 625
#define BATCH           32
#define LDS_DW          8192                     // 32 KB staging window -> 10 blocks/WGP

typedef unsigned int v4u __attribute__((ext_vector_type(4)));
typedef int          v8i __attribute__((ext_vector_type(8)));
typedef int          v4i __attribute__((ext_vector_type(4)));

__device__ __forceinline__ int iclamp(int v, int lo, int hi) {
    return v < lo ? lo : (v > hi ? hi : v);
}

__global__ __launch_bounds__(256) void stn_bilinear_kernel(
    const float* __restrict__ img,     // [32,512,512,3]
    const float* __restrict__ theta,   // [32,2,3]
    float* __restrict__ out)           // [32,400,400,3]
{
    __shared__ float tile[LDS_DW];

    const int blk = blockIdx.x;
    const int b   = blk / TILES_PER_BATCH;            // uniform per block
    const int t   = blk - b * TILES_PER_BATCH;
    const int tx  = t % TILES_PER_SIDE;
    const int ty  = t / TILES_PER_SIDE;

    // theta is uniform per block -> scalar loads
    const float* th = theta + b * 6;
    const float t00 = th[0], t01 = th[1], t02 = th[2];
    const float t10 = th[3], t11 = th[4], t12 = th[5];

    const float step = 2.0f / (float)(OUT_HW - 1);    // linspace(-1,1,400)

    // ---- input-space bounding box of this output tile (affine -> extremes at corners) ----
    const float gxa = -1.0f + step * (float)(tx * TILE);
    const float gxb = -1.0f + step * (float)(tx * TILE + TILE - 1);
    const float gya = -1.0f + step * (float)(ty * TILE);
    const float gyb = -1.0f + step * (float)(ty * TILE + TILE - 1);

    float cx[4], cy[4];
    {
        const float gxs[4] = {gxa, gxb, gxa, gxb};
        const float gys[4] = {gya, gya, gyb, gyb};
#pragma unroll
        for (int i = 0; i < 4; ++i) {
            cx[i] = 0.5f * (fmaf(t00, gxs[i], fmaf(t01, gys[i], t02)) + 1.0f) * (float)IMG_HW;
            cy[i] = 0.5f * (fmaf(t10, gxs[i], fmaf(t11, gys[i], t12)) + 1.0f) * (float)IMG_HW;
        }
    }
    float xmin = fminf(fminf(cx[0], cx[1]), fminf(cx[2], cx[3]));
    float xmax = fmaxf(fmaxf(cx[0], cx[1]), fmaxf(cx[2], cx[3]));
    float ymin = fminf(fminf(cy[0], cy[1]), fminf(cy[2], cy[3]));
    float ymax = fmaxf(fmaxf(cy[0], cy[1]), fmaxf(cy[2], cy[3]));
    xmin = fmaxf(xmin, -1.0e6f); xmax = fminf(xmax, 1.0e6f);
    ymin = fmaxf(ymin, -1.0e6f); ymax = fminf(ymax, 1.0e6f);

    const int xlo  = iclamp((int)floorf(xmin),     0, IMG_HW - 1);
    const int xhi  = iclamp((int)floorf(xmax) + 1, 0, IMG_HW - 1);
    const int ylo  = iclamp((int)floorf(ymin),     0, IMG_HW - 1);
    const int yhi  = iclamp((int)floorf(ymax) + 1, 0, IMG_HW - 1);

    const int xloD = (xlo * 3) & ~3;                  // 16B-aligned window start (dwords)
    const int xhiD = ((xhi * 3 + 2) | 3) + 1;         // exclusive, multiple of 4, <= 1536
    const int Wd   = xhiD - xloD;                     // staged dwords per row
    const int Hb   = yhi - ylo + 1;                   // staged rows
    const bool useLds = (Wd * Hb) <= LDS_DW;          // uniform per block

    const float* gwin = img + (size_t)b * IMG_DW + (size_t)ylo * ROW_DW + xloD;

    if (useLds) {
#if __has_builtin(__builtin_amdgcn_tensor_load_to_lds) && __has_builtin(__builtin_amdgcn_s_wait_tensorcnt)
        // One TDM descriptor copies the whole 2D window global->LDS (wave 0 issues once).
        if (threadIdx.x < 32) {
            const uint64_t ga = (uint64_t)(uintptr_t)gwin;
            const uint32_t la = (uint32_t)(uintptr_t)(&tile[0]);   // low 32 bits = LDS byte offset

            v4u g0;
            g0[0] = (unsigned)__builtin_amdgcn_readfirstlane((int)1u);              // count=1 (valid user D#)
            g0[1] = (unsigned)__builtin_amdgcn_readfirstlane((int)la);              // lds_addr (bytes)
            g0[2] = (unsigned)__builtin_amdgcn_readfirstlane((int)(uint32_t)ga);    // global_addr[31:0]
            g0[3] = (unsigned)__builtin_amdgcn_readfirstlane(
                        (int)(((uint32_t)(ga >> 32) & 0x01FFFFFFu) | (2u << 30)));  // addr[56:32] | type=2

            v8i g1;
            g1[0] = __builtin_amdgcn_readfirstlane((int)(2u << 16));        // data_size=4B, no mask/pad/iterate
            g1[1] = __builtin_amdgcn_readfirstlane((int)0xFFFF0000u);       // tensor_dim0 = 0x7FFFFFFF (lo16)
            g1[2] = __builtin_amdgcn_readfirstlane((int)0xFFFF7FFFu);       // dim0 hi16 | tensor_dim1 lo16
            g1[3] = __builtin_amdgcn_readfirstlane((int)(0x7FFFu | ((unsigned)Wd << 16))); // dim1 hi16 | tile_dim0
            g1[4] = __builtin_amdgcn_readfirstlane(Hb);                     // tile_dim1=Hb, tile_dim2=0 (2D)
            g1[5] = __builtin_amdgcn_readfirstlane(ROW_DW);                 // tensor_dim0_stride = 1536
            g1[6] = 0;
            g1[7] = 0;

            v4i gz = {0, 0, 0, 0};
#if __clang_major__ >= 23
            v8i gz8 = {0, 0, 0, 0, 0, 0, 0, 0};
            __builtin_amdgcn_tensor_load_to_lds(g0, g1, gz, gz, gz8, 0);
#else
            __builtin_amdgcn_tensor_load_to_lds(g0, g1, gz, gz, 0);
#endif
            __builtin_amdgcn_s_wait_tensorcnt(0);
        }
#else
        // Sync staging fallback: wave r-strided rows, lane c-strided dwords.
        {
            const int lane = threadIdx.x & 31;
            const int wv   = threadIdx.x >> 5;
            for (int r = wv; r < Hb; r += 8)
                for (int c = lane; c < Wd; c += 32)
                    tile[r * Wd + c] = gwin[r * ROW_DW + c];
        }
#endif
        __syncthreads();
    }

    // ---------------- per-pixel bilinear sample ----------------
    const int lx = threadIdx.x & (TILE - 1);
    const int ly = threadIdx.x >> 4;
    const int ox = tx * TILE + lx;
    const int oy = ty * TILE + ly;

    const float gx = -1.0f + step * (float)ox;
    const float gy = -1.0f + step * (float)oy;
    const float x = 0.5f * (fmaf(t00, gx, fmaf(t01, gy, t02)) + 1.0f) * (float)IMG_HW;
    const float y = 0.5f * (fmaf(t10, gx, fmaf(t11, gy, t12)) + 1.0f) * (float)IMG_HW;

    const int x0 = (int)floorf(x);
    const int y0 = (int)floorf(y);
    const int x0c = iclamp(x0,     0, IMG_HW - 1);
    const int x1c = iclamp(x0 + 1, 0, IMG_HW - 1);
    const int y0c = iclamp(y0,     0, IMG_HW - 1);
    const int y1c = iclamp(y0 + 1, 0, IMG_HW - 1);

    // weights use the *clamped* integer coords, exactly like the reference
    const float wxa = (float)x1c - x;
    const float wxb = x - (float)x0c;
    const float wya = (float)y1c - y;
    const float wyb = y - (float)y0c;
    const float wa = wxa * wya, wb = wxa * wyb, wc = wxb * wya, wd = wxb * wyb;

    float r0, r1, r2;
    if (useLds) {
        const int ca = x0c * 3 - xloD;
        const int cb = x1c * 3 - xloD;
        const int ra = (y0c - ylo) * Wd;
        const int rb = (y1c - ylo) * Wd;
        const float* pa = &tile[ra + ca];
        const float* pb = &tile[rb + ca];
        const float* pc = &tile[ra + cb];
        const float* pd = &tile[rb + cb];
        r0 = fmaf(wa, pa[0], fmaf(wb, pb[0], fmaf(wc, pc[0], wd * pd[0])));
        r1 = fmaf(wa, pa[1], fmaf(wb, pb[1], fmaf(wc, pc[1], wd * pd[1])));
        r2 = fmaf(wa, pa[2], fmaf(wb, pb[2], fmaf(wc, pc[2], wd * pd[2])));
    } else {
        const float* ib = img + (size_t)b * IMG_DW;
        const float* pa = ib + (size_t)(y0c * IMG_HW + x0c) * 3;
        const float* pb = ib + (size_t)(y1c * IMG_HW + x0c) * 3;
        const float* pc = ib + (size_t)(y0c * IMG_HW + x1c) * 3;
        const float* pd = ib + (size_t)(y1c * IMG_HW + x1c) * 3;
        r0 = fmaf(wa, pa[0], fmaf(wb, pb[0], fmaf(wc, pc[0], wd * pd[0])));
        r1 = fmaf(wa, pa[1], fmaf(wb, pb[1], fmaf(wc, pc[1], wd * pd[1])));
        r2 = fmaf(wa, pa[2], fmaf(wb, pb[2], fmaf(wc, pc[2], wd * pd[2])));
    }

    float* o = out + (size_t)((b * OUT_HW + oy) * OUT_HW + ox) * 3;
    o[0] = r0;
    o[1] = r1;
    o[2] = r2;
}

extern "C" void kernel_launch(void* const* d_in, const int* in_sizes, int n_in,
                              void* d_out, int out_size, void* d_ws, size_t ws_size,
                              hipStream_t stream) {
    (void)in_sizes; (void)n_in; (void)out_size; (void)d_ws; (void)ws_size;
    const float* img   = (const float*)d_in[0];
    const float* theta = (const float*)d_in[1];
    float* out         = (float*)d_out;

    dim3 grid(BATCH * TILES_PER_BATCH);   // 32 * 625 = 20000 blocks
    dim3 block(256);                      // 8 wave32s, one 16x16 output tile
    stn_bilinear_kernel<<<grid, block, 0, stream>>>(img, theta, out);
}